// InterViews_18674517803185
// MI455X (gfx1250) — compile-verified
//
#include <hip/hip_runtime.h>

typedef __attribute__((ext_vector_type(2))) float v2f;
typedef __attribute__((ext_vector_type(8))) float v8f;

#define BS 4096   // batch positions
#define CH 2048   // channels
#define NV 8      // views

// One wave32 computes the 16x16 fp32 Gram of the 8 view-vectors of TWO batch
// items (rows 0-7 = views of b0, rows 8-15 = views of b1) using
// V_WMMA_F32_16X16X4_F32, accumulating over K = 2048 in steps of 4.
//
// For V_WMMA_F32_16X16X4_F32 (wave32):
//   A 16x4:  lane l -> M = l&15;  VGPR j holds K = j + 2*(l>>4)
//   B 4x16:  lane l -> N = l&15;  VGPR j holds K = j + 2*(l>>4)
// For a Gram matrix (B == A^T of the same data) these are the SAME register
// image, so one float2 load per lane feeds both SRC0 and SRC1.
__global__ __launch_bounds__(256) void interviews_gram_kernel(
    const float* __restrict__ X, float* __restrict__ out) {
  const int lane = threadIdx.x & 31;
  const int wave = threadIdx.x >> 5;
  const int half = lane >> 4;   // which K half this lane supplies
  const int m    = lane & 15;   // M (for A) == N (for B)
  const int view = m & 7;
  const int bsel = m >> 3;      // 0 -> b0, 1 -> b1
  const int bpair = blockIdx.x * 8 + wave;     // 0..2047
  const int b     = bpair * 2 + bsel;

  // row (view, b) of vision_features, element offset 2*half within each K-step
  const float* src = X + (size_t)(view * BS + b) * CH + 2 * half;

  v8f acc = {};
#pragma unroll 8
  for (int k = 0; k < CH; k += 4) {
    // 8B non-temporal load: elements [k + 2*half, k + 2*half + 1]
    v2f a = __builtin_nontemporal_load((const v2f*)(src + k));
    // D = A*B + C ; A == B == this wave's 16x4 tile (Gram)
    acc = __builtin_amdgcn_wmma_f32_16x16x4_f32(
        /*neg_a=*/false, a, /*neg_b=*/false, a,
        /*c_mod=*/(short)0, acc, /*reuse_a=*/false, /*reuse_b=*/false);
  }

  // C/D layout: VGPR v, lanes 0-15 -> (M=v,   N=lane),
  //                     lanes 16-31 -> (M=v+8, N=lane-16).
  // => G(b0) (M,N in 0..7)  lives in lanes 0..7,
  //    G(b1) (M,N in 8..15) lives in lanes 24..31.
  __shared__ alignas(32) float lds[8][32][8];
  *(v8f*)(&lds[wave][lane][0]) = acc;
  __syncthreads();

  if (lane < 2) {
    const int base = lane ? 24 : 0;  // lane0 -> b0 block, lane1 -> b1 block
    float Gm[8][8];  // Gm[n][v] = G[v][n] (symmetric, so orientation is moot)
#pragma unroll
    for (int n = 0; n < 8; ++n) {
      v8f r = *(const v8f*)(&lds[wave][base + n][0]);
#pragma unroll
      for (int v = 0; v < 8; ++v) Gm[n][v] = r[v];
    }
    float inv[8];
#pragma unroll
    for (int v = 0; v < 8; ++v) inv[v] = 1.0f / sqrtf(Gm[v][v]);

    float sum = 0.0f, sumsq = 0.0f;
#pragma unroll
    for (int v = 0; v < 8; ++v) {
#pragma unroll
      for (int w = v + 1; w < 8; ++w) {
        float s = Gm[v][w] * inv[v] * inv[w];
        sum += s;
        sumsq += s * s;
      }
    }
    const float mean = sum * (1.0f / 28.0f);
    const float var  = (sumsq - 28.0f * mean * mean) * (1.0f / 27.0f);
    out[bpair * 2 + lane] = -var;
  }
}

extern "C" void kernel_launch(void* const* d_in, const int* in_sizes, int n_in,
                              void* d_out, int out_size, void* d_ws, size_t ws_size,
                              hipStream_t stream) {
  const float* X = (const float*)d_in[0];  // (8*4096, 2048) fp32
  float* out = (float*)d_out;              // (4096,) fp32
  (void)in_sizes; (void)n_in; (void)out_size; (void)d_ws; (void)ws_size;

  dim3 grid(BS / 16);   // 256 blocks: 8 waves/block * 2 batch items/wave
  dim3 block(256);
  hipLaunchKernelGGL(interviews_gram_kernel, grid, block, 0, stream, X, out);
}